// UncertaintyAwareAggregator_80582176407864
// MI455X (gfx1250) — compile-verified
//
#include <hip/hip_runtime.h>
#include <hip/hip_bf16.h>

#define DIM 2048
#define HH 256
#define N_HEADS 4
#define HEAD_DIM 64
#define N_SIG 6
#define SW_HID 4096
#define BB 4
#define SS 2048
#define EPSV 1e-6f
#define KSPLIT 4

typedef __attribute__((ext_vector_type(2))) float v2f;
typedef __attribute__((ext_vector_type(8))) float v8f;

// ---------------------------------------------------------------------------
// Kernel 1: tiny head — signals, confidence, 6-token attention, unc_summary.
// One block, 256 threads. Writes padded u [16 x 256] (rows 4..15 zero).
// ---------------------------------------------------------------------------
__global__ __launch_bounds__(256) void head_kernel(
    const float* __restrict__ epi, const float* __restrict__ ale,
    const float* __restrict__ ood, const float* __restrict__ cu,
    const float* __restrict__ ece, const float* __restrict__ moral,
    const float* __restrict__ sp_w, const float* __restrict__ sp_b,
    const float* __restrict__ in_w, const float* __restrict__ in_b,
    const float* __restrict__ out_w, const float* __restrict__ out_b,
    const float* __restrict__ ch_w1, const float* __restrict__ ch_b1,
    const float* __restrict__ ch_w2, const float* __restrict__ ch_b2,
    float* __restrict__ sig_out, float* __restrict__ conf_out,
    float* __restrict__ u_out) {
  __shared__ float s_sig[BB][N_SIG];
  __shared__ float s_red[256];
  __shared__ float s_A[3 * HH];   // in_proj_w[j] . sp_w
  __shared__ float s_C[3 * HH];   // in_proj_w[j] . sp_b + in_proj_b[j]
  __shared__ float s_qkv[N_SIG][3 * HH];
  __shared__ float s_attn[N_HEADS][N_SIG][N_SIG];
  __shared__ float s_obar[BB][HH];
  const int t = threadIdx.x;

  // ---- signals [B,6] ----
  if (t < BB * N_SIG) {
    int b = t / N_SIG, j = t % N_SIG;
    float v;
    if (j == 0) { float s = 0.f; for (int k = 0; k < 64; ++k) s += epi[b * 64 + k]; v = s * (1.f / 64.f); }
    else if (j == 1) { float s = 0.f; for (int k = 0; k < 64; ++k) s += ale[b * 64 + k]; v = s * (1.f / 64.f); }
    else if (j == 2) v = ood[b];
    else if (j == 3) v = cu[b];
    else if (j == 4) v = ece[0];
    else v = moral[b];
    s_sig[b][j] = v;
    sig_out[b * N_SIG + j] = v;
  }
  __syncthreads();

  // ---- confidence head: GELU(exact) + sigmoid ----
  for (int b = 0; b < BB; ++b) {
    float z = ch_b1[t];
    #pragma unroll
    for (int k = 0; k < N_SIG; ++k) z += s_sig[b][k] * ch_w1[t * N_SIG + k];
    float g = 0.5f * z * (1.0f + erff(z * 0.70710678118654752f));
    s_red[t] = g * ch_w2[t];
    __syncthreads();
    for (int s = 128; s > 0; s >>= 1) { if (t < s) s_red[t] += s_red[t + s]; __syncthreads(); }
    if (t == 0) {
      float v = s_red[0] + ch_b2[0];
      conf_out[b] = 1.0f / (1.0f + __expf(-v));
    }
    __syncthreads();
  }

  // ---- fold signal_proj into qkv: qkv[b][i][j] = sig*A_j + C_j ----
  for (int j = t; j < 3 * HH; j += 256) {
    float a = 0.f, c = 0.f;
    for (int h = 0; h < HH; ++h) {
      float w = in_w[j * HH + h];
      a += w * sp_w[h];
      c += w * sp_b[h];
    }
    s_A[j] = a;
    s_C[j] = c + in_b[j];
  }
  __syncthreads();

  for (int b = 0; b < BB; ++b) {
    for (int idx = t; idx < N_SIG * 3 * HH; idx += 256) {
      int i = idx / (3 * HH), j = idx % (3 * HH);
      s_qkv[i][j] = s_sig[b][i] * s_A[j] + s_C[j];
    }
    __syncthreads();
    // scores + softmax per (head, query-row)
    if (t < N_HEADS * N_SIG) {
      int h = t / N_SIG, i = t % N_SIG;
      float sc[N_SIG];
      float m = -1e30f;
      for (int j = 0; j < N_SIG; ++j) {
        float s = 0.f;
        for (int d = 0; d < HEAD_DIM; ++d)
          s += s_qkv[i][h * HEAD_DIM + d] * s_qkv[j][HH + h * HEAD_DIM + d];
        sc[j] = s * 0.125f;  // 1/sqrt(64)
        m = fmaxf(m, sc[j]);
      }
      float den = 0.f;
      for (int j = 0; j < N_SIG; ++j) { sc[j] = __expf(sc[j] - m); den += sc[j]; }
      float inv = 1.0f / den;
      for (int j = 0; j < N_SIG; ++j) s_attn[h][i][j] = sc[j] * inv;
    }
    __syncthreads();
    // obar[b][hd] = mean_i sum_j attn * v
    {
      int head = t >> 6;
      float acc = 0.f;
      for (int i = 0; i < N_SIG; ++i)
        for (int j = 0; j < N_SIG; ++j)
          acc += s_attn[head][i][j] * s_qkv[j][2 * HH + t];
      s_obar[b][t] = acc * (1.0f / 6.0f);
    }
    __syncthreads();
  }

  // ---- out proj + padded u [16 x 256] ----
  for (int b = 0; b < BB; ++b) {
    float acc = out_b[t];
    for (int h = 0; h < HH; ++h) acc += s_obar[b][h] * out_w[t * HH + h];
    u_out[b * HH + t] = acc;
  }
  for (int r = BB; r < 16; ++r) u_out[r * HH + t] = 0.0f;
}

// ---------------------------------------------------------------------------
// Split-K WMMA f32 16x16x4 GEMM: C[16,N] = A[16,K] @ W[N,K]^T (+bias).
// 4 waves per block, each covering K/4; partial 16x16 tiles reduced in LDS.
// A/B layout per ISA 7.12.2 (32-bit 16x4): lane<16 holds K={k0,k0+1},
// lane>=16 holds K={k0+2,k0+3} of row M/N = lane%16 -> contiguous float2
// per lane; lanes l and l+16 stream adjacent pairs of the same weight row,
// so each weight cache line is fully consumed within the wave's K chunk.
// ---------------------------------------------------------------------------
__global__ __launch_bounds__(32 * KSPLIT) void wmma_gemm16(
    const float* __restrict__ A, const float* __restrict__ W,
    const float* __restrict__ bias, float* __restrict__ C, int K, int N) {
  __shared__ float s_part[KSPLIT][256];
  const int lane = threadIdx.x & 31;
  const int wave = threadIdx.x >> 5;
  const int half = lane >> 4;
  const int r16 = lane & 15;
  const int n0 = blockIdx.x * 16;
  const int Kc = K / KSPLIT;
  const float* ap = A + (size_t)r16 * K + wave * Kc + 2 * half;
  const float* wp = W + (size_t)(n0 + r16) * K + wave * Kc + 2 * half;
  v8f acc = {};
  for (int k0 = 0; k0 < Kc; k0 += 4) {
    __builtin_prefetch(wp + k0 + 512, 0, 3);  // WGP-scope: lines re-hit over K
    v2f a = *(const v2f*)(ap + k0);
    v2f b = *(const v2f*)(wp + k0);
    acc = __builtin_amdgcn_wmma_f32_16x16x4_f32(false, a, false, b, (short)0,
                                                acc, false, false);
  }
  #pragma unroll
  for (int r = 0; r < 8; ++r) s_part[wave][lane * 8 + r] = acc[r];
  __syncthreads();
  // reduce partials: 128 threads x 2 elements = 256 tile values
  #pragma unroll
  for (int e = 0; e < 2; ++e) {
    int idx = threadIdx.x + e * 128;
    float s = s_part[0][idx] + s_part[1][idx] + s_part[2][idx] + s_part[3][idx];
    int l = idx >> 3, r = idx & 7;
    int m = r + 8 * (l >> 4);
    int n = n0 + (l & 15);
    if (bias) s += bias[n];
    C[(size_t)m * N + n] = s;
  }
}

// ---------------------------------------------------------------------------
// Split-K WMMA SwiGLU: H[16,N] = silu(A @ W1^T) * (A @ W3^T).
// ---------------------------------------------------------------------------
__global__ __launch_bounds__(32 * KSPLIT) void wmma_swiglu16(
    const float* __restrict__ A, const float* __restrict__ W1,
    const float* __restrict__ W3, float* __restrict__ H, int K, int N) {
  __shared__ float s_p1[KSPLIT][256];
  __shared__ float s_p3[KSPLIT][256];
  const int lane = threadIdx.x & 31;
  const int wave = threadIdx.x >> 5;
  const int half = lane >> 4;
  const int r16 = lane & 15;
  const int n0 = blockIdx.x * 16;
  const int Kc = K / KSPLIT;
  const float* ap = A + (size_t)r16 * K + wave * Kc + 2 * half;
  const float* w1p = W1 + (size_t)(n0 + r16) * K + wave * Kc + 2 * half;
  const float* w3p = W3 + (size_t)(n0 + r16) * K + wave * Kc + 2 * half;
  v8f acc1 = {}, acc3 = {};
  for (int k0 = 0; k0 < Kc; k0 += 4) {
    __builtin_prefetch(w1p + k0 + 512, 0, 3);
    __builtin_prefetch(w3p + k0 + 512, 0, 3);
    v2f a = *(const v2f*)(ap + k0);
    v2f b1 = *(const v2f*)(w1p + k0);
    v2f b3 = *(const v2f*)(w3p + k0);
    acc1 = __builtin_amdgcn_wmma_f32_16x16x4_f32(false, a, false, b1, (short)0,
                                                 acc1, false, false);
    acc3 = __builtin_amdgcn_wmma_f32_16x16x4_f32(false, a, false, b3, (short)0,
                                                 acc3, false, false);
  }
  #pragma unroll
  for (int r = 0; r < 8; ++r) {
    s_p1[wave][lane * 8 + r] = acc1[r];
    s_p3[wave][lane * 8 + r] = acc3[r];
  }
  __syncthreads();
  #pragma unroll
  for (int e = 0; e < 2; ++e) {
    int idx = threadIdx.x + e * 128;
    float g = s_p1[0][idx] + s_p1[1][idx] + s_p1[2][idx] + s_p1[3][idx];
    float v3 = s_p3[0][idx] + s_p3[1][idx] + s_p3[2][idx] + s_p3[3][idx];
    float sig = g / (1.0f + __expf(-g));  // silu
    int l = idx >> 3, r = idx & 7;
    int m = r + 8 * (l >> 4);
    int n = n0 + (l & 15);
    H[(size_t)m * N + n] = sig * v3;
  }
}

// ---------------------------------------------------------------------------
// In-place row RMSNorm over 2048 cols, one block per row.
// ---------------------------------------------------------------------------
__global__ __launch_bounds__(256) void rmsnorm_rows(
    float* __restrict__ T, const float* __restrict__ w) {
  __shared__ float red[256];
  const int t = threadIdx.x;
  float* row = T + (size_t)blockIdx.x * DIM;
  float v[8];
  float ss = 0.f;
  #pragma unroll
  for (int i = 0; i < 8; ++i) { v[i] = row[t + 256 * i]; ss += v[i] * v[i]; }
  red[t] = ss;
  __syncthreads();
  for (int s = 128; s > 0; s >>= 1) { if (t < s) red[t] += red[t + s]; __syncthreads(); }
  const float scale = rsqrtf(red[0] * (1.0f / DIM) + EPSV);
  #pragma unroll
  for (int i = 0; i < 8; ++i) row[t + 256 * i] = v[i] * scale * w[t + 256 * i];
}

// ---------------------------------------------------------------------------
// Fused writer: x_aware = rmsnorm(x + 0.05*emb[b], norm_w); also broadcasts
// unc_embedding rows. One block per (b,s) row; float4 streams; pure BW.
// ---------------------------------------------------------------------------
__global__ __launch_bounds__(256) void fuse_rows(
    const float* __restrict__ x, const float* __restrict__ emb,
    const float* __restrict__ nw, float* __restrict__ xa,
    float* __restrict__ embo) {
  __shared__ float red[256];
  const int t = threadIdx.x;
  const size_t r = blockIdx.x;
  const int b = (int)(r >> 11);  // r / 2048
  const float4* xr = (const float4*)(x + r * DIM);
  const float4* er = (const float4*)(emb + (size_t)b * DIM);
  const float4* wr = (const float4*)nw;
  float4* xo = (float4*)(xa + r * DIM);
  float4* eo = (float4*)(embo + r * DIM);

  float4 e0 = er[t], e1 = er[t + 256];
  float4 x0 = xr[t], x1 = xr[t + 256];
  float4 y0, y1;
  y0.x = x0.x + 0.05f * e0.x; y0.y = x0.y + 0.05f * e0.y;
  y0.z = x0.z + 0.05f * e0.z; y0.w = x0.w + 0.05f * e0.w;
  y1.x = x1.x + 0.05f * e1.x; y1.y = x1.y + 0.05f * e1.y;
  y1.z = x1.z + 0.05f * e1.z; y1.w = x1.w + 0.05f * e1.w;

  float ss = y0.x * y0.x + y0.y * y0.y + y0.z * y0.z + y0.w * y0.w +
             y1.x * y1.x + y1.y * y1.y + y1.z * y1.z + y1.w * y1.w;
  red[t] = ss;
  __syncthreads();
  for (int s = 128; s > 0; s >>= 1) { if (t < s) red[t] += red[t + s]; __syncthreads(); }
  const float scale = rsqrtf(red[0] * (1.0f / DIM) + EPSV);

  float4 w0 = wr[t], w1 = wr[t + 256];
  float4 o0, o1;
  o0.x = y0.x * scale * w0.x; o0.y = y0.y * scale * w0.y;
  o0.z = y0.z * scale * w0.z; o0.w = y0.w * scale * w0.w;
  o1.x = y1.x * scale * w1.x; o1.y = y1.y * scale * w1.y;
  o1.z = y1.z * scale * w1.z; o1.w = y1.w * scale * w1.w;
  xo[t] = o0; xo[t + 256] = o1;
  eo[t] = e0; eo[t + 256] = e1;
}

// ---------------------------------------------------------------------------
extern "C" void kernel_launch(void* const* d_in, const int* in_sizes, int n_in,
                              void* d_out, int out_size, void* d_ws, size_t ws_size,
                              hipStream_t stream) {
  const float* x      = (const float*)d_in[0];
  const float* epi    = (const float*)d_in[1];
  const float* ale    = (const float*)d_in[2];
  const float* ood    = (const float*)d_in[3];
  const float* cu     = (const float*)d_in[4];
  const float* ece    = (const float*)d_in[5];
  const float* moral  = (const float*)d_in[6];
  const float* sp_w   = (const float*)d_in[7];
  const float* sp_b   = (const float*)d_in[8];
  const float* in_w   = (const float*)d_in[9];
  const float* in_b   = (const float*)d_in[10];
  const float* out_w  = (const float*)d_in[11];
  const float* out_b  = (const float*)d_in[12];
  const float* ch_w1  = (const float*)d_in[13];
  const float* ch_b1  = (const float*)d_in[14];
  const float* ch_w2  = (const float*)d_in[15];
  const float* ch_b2  = (const float*)d_in[16];
  const float* sum_w1 = (const float*)d_in[17];
  const float* sum_b1 = (const float*)d_in[18];
  const float* rms1_w = (const float*)d_in[19];
  const float* sw_w1  = (const float*)d_in[20];
  const float* sw_w3  = (const float*)d_in[21];
  const float* sw_w2  = (const float*)d_in[22];
  const float* sum_w2 = (const float*)d_in[23];
  const float* sum_b2 = (const float*)d_in[24];
  const float* rms2_w = (const float*)d_in[25];
  const float* norm_w = (const float*)d_in[26];

  float* ws = (float*)d_ws;
  float* u  = ws;                    // 16*256   = 4096
  float* t  = ws + 4096;             // 16*2048  = 32768
  float* h  = ws + 36864;            // 16*4096  = 65536
  float* t2 = ws + 102400;           // 16*2048
  float* t3 = ws + 135168;           // 16*2048 (becomes unc_emb rows)

  float* xa   = (float*)d_out;                     // [B,S,DIM]
  float* conf = xa + (size_t)BB * SS * DIM;        // [B]
  float* sig  = conf + BB;                         // [B,6]
  float* embo = sig + BB * N_SIG;                  // [B,S,DIM]

  head_kernel<<<1, 256, 0, stream>>>(epi, ale, ood, cu, ece, moral, sp_w, sp_b,
                                     in_w, in_b, out_w, out_b, ch_w1, ch_b1,
                                     ch_w2, ch_b2, sig, conf, u);
  wmma_gemm16<<<DIM / 16, 32 * KSPLIT, 0, stream>>>(u, sum_w1, sum_b1, t, HH, DIM);
  rmsnorm_rows<<<16, 256, 0, stream>>>(t, rms1_w);
  wmma_swiglu16<<<SW_HID / 16, 32 * KSPLIT, 0, stream>>>(t, sw_w1, sw_w3, h, DIM, SW_HID);
  wmma_gemm16<<<DIM / 16, 32 * KSPLIT, 0, stream>>>(h, sw_w2, nullptr, t2, SW_HID, DIM);
  wmma_gemm16<<<DIM / 16, 32 * KSPLIT, 0, stream>>>(t2, sum_w2, sum_b2, t3, DIM, DIM);
  rmsnorm_rows<<<16, 256, 0, stream>>>(t3, rms2_w);
  fuse_rows<<<BB * SS, 256, 0, stream>>>(x, t3, norm_w, xa, embo);
}